// NoHybridANFIS_29978871726294
// MI455X (gfx1250) — compile-verified
//
#include <hip/hip_runtime.h>
#include <hip/hip_bf16.h>

// Problem constants (match reference)
constexpr int B = 4096, D = 16, M = 4, R = 2048, C = 10;
constexpr int DM = D * M;            // 64 = K dim of the selector GEMM
constexpr int CE = 16;               // padded consequent columns (10 data + col10=ones)
constexpr int NTILE = R / 16;        // 128 rule tiles
constexpr float EPS = 1e-9f;
constexpr float HALF_LOG2E = 0.7213475204444817f;  // log2(e)/2

constexpr int SELPACK_FLOATS = NTILE * 16 * 32 * 2;   // 131072 (512 KB)
constexpr int SCONS_FLOATS   = R * CE;                // 32768  (128 KB)

typedef __attribute__((ext_vector_type(2))) float v2f;
typedef __attribute__((ext_vector_type(8))) float v8f;

// ---------------------------------------------------------------------------
// Kernel 1 (prep):
//   - sel_pack: one-hot selector S[k,r], pre-swizzled into the exact WMMA
//     B-fragment layout (tile, k-slice, lane, elem) so the main kernel does
//     pure v2f loads instead of per-tile cmp/cndmask chains.
//   - scons_ext[r][0..9] = sum_j consequents[r][j][c]; [r][10] = 1.0 (row-sum
//     column); [r][11..15] = 0.
//   - x_ext output (x with bias column).
// ---------------------------------------------------------------------------
__global__ void anfis_prep_kernel(const float* __restrict__ x,
                                  const float* __restrict__ consequents,
                                  const int*   __restrict__ rules,
                                  float* __restrict__ sel_pack,
                                  float* __restrict__ scons_ext,
                                  float* __restrict__ x_ext) {
    const int id = blockIdx.x * blockDim.x + threadIdx.x;

    if (id < SELPACK_FLOATS) {
        const int v    = id & 1;
        const int lane = (id >> 1) & 31;
        const int ks   = (id >> 6) & 15;          // k-slice == input dim d
        const int rb   = id >> 10;                // rule tile
        const int r    = rb * 16 + (lane & 15);   // B-fragment column N
        const int m    = (lane >> 4) * 2 + v;     // membership index
        sel_pack[id] = (rules[r * D + ks] == m) ? 1.0f : 0.0f;
    }
    if (id < SCONS_FLOATS) {
        const int r = id >> 4, cc = id & 15;
        float s;
        if (cc < C) {
            s = 0.0f;
#pragma unroll
            for (int j = 0; j < D + 1; ++j)
                s += consequents[(r * (D + 1) + j) * C + cc];
        } else {
            s = (cc == C) ? 1.0f : 0.0f;          // ones column -> row sums
        }
        scons_ext[id] = s;
    }
    if (id < B * (D + 1)) {
        const int b = id / (D + 1), j = id % (D + 1);
        x_ext[id] = (j < D) ? x[b * D + j] : 1.0f;
    }
}

// ---------------------------------------------------------------------------
// Kernel 2: fused main kernel.
//   GEMM1 (WMMA f32 16x16x4, K=64): log2(fs) = logmf @ one-hot-selector
//   GEMM2 (WMMA f32 16x16x4, K=16/tile, chained over 8 tiles):
//          out_un(+rowsum col) = fs @ scons_ext  (fs transposed via LDS)
//   Then normalize fs in registers and store norm_fs exactly once.
// Workgroup = 16 batch rows x all 2048 rules; 512 threads = 16 wave32,
// each wave owns 8 rule tiles of 16.
// ---------------------------------------------------------------------------
__global__ __launch_bounds__(512, 1)
void anfis_main_kernel(const float* __restrict__ x,
                       const float* __restrict__ centers,
                       const float* __restrict__ widths,
                       const float* __restrict__ sel_pack,
                       const float* __restrict__ scons_ext,
                       float* __restrict__ out,
                       float* __restrict__ norm_fs) {
    __shared__ float lmf[16][DM];        // log2-domain memberships (16 rows)
    __shared__ float out_un[16][CE];     // accumulated fs @ scons_ext
    __shared__ float sxs[16];            // sum_i x_ext[b,i]
    __shared__ float ltr[16][16 * 17];   // per-wave fs transpose tile (pad 17)

    const int tid   = threadIdx.x;
    const int bBase = blockIdx.x * 16;

    // ---- Phase 0: logmf tile in LDS (scaled so exp2 finishes it) ----------
    for (int i = tid; i < 16 * DM; i += 512) {
        const int row = i >> 6;
        const int dm  = i & 63;
        const int d   = dm >> 2;
        const int m   = dm & 3;
        const float xv = x[(bBase + row) * D + d];
        const float cc = centers[d * M + m];
        const float w  = widths[d * M + m];
        const float diff = xv - cc;
        lmf[row][dm] = -diff * diff * (HALF_LOG2E / (w * w));
    }
    if (tid < 16) {
        float s = 1.0f;  // bias column
#pragma unroll
        for (int d = 0; d < D; ++d) s += x[(bBase + tid) * D + d];
        sxs[tid] = s;
    }
    if (tid < 16 * CE) out_un[tid / CE][tid % CE] = 0.0f;
    __syncthreads();

    const int wave   = tid >> 5;
    const int lane   = tid & 31;
    const int laneLo = lane & 15;        // A row M / B col N / D col N
    const int laneHi = lane >> 4;        // K-group selector

    // A fragments of GEMM1: lanes 0-15 hold K={0,1}, lanes 16-31 K={2,3}
    v2f a[16];
#pragma unroll
    for (int ks = 0; ks < 16; ++ks) {
        a[ks].x = lmf[laneLo][ks * 4 + laneHi * 2 + 0];
        a[ks].y = lmf[laneLo][ks * 4 + laneHi * 2 + 1];
    }

    float fs[8][8];                      // [tile][acc-vgpr] firing strengths
    v8f oacc = {};                       // GEMM2 accumulator (16m x 16c tile)
    float* tr = &ltr[wave][0];

#pragma unroll
    for (int t = 0; t < 8; ++t) {
        const int rb    = wave * 8 + t;          // global rule tile index
        const int rBase = rb * 16;

        // ---- GEMM1: log2(fs) tile = logmf @ selector (16 K-slices) --------
        const v2f* selp =
            reinterpret_cast<const v2f*>(sel_pack + (size_t)rb * 16 * 64);
        v8f acc = {};
#pragma unroll
        for (int ks = 0; ks < 16; ++ks) {
            const v2f bf = selp[ks * 32 + lane];  // pre-swizzled B fragment
            acc = __builtin_amdgcn_wmma_f32_16x16x4_f32(
                false, a[ks], false, bf, (short)0, acc, false, false);
        }

        // ---- exp2 -> fs; stage transposed copy in LDS for GEMM2 -----------
#pragma unroll
        for (int v = 0; v < 8; ++v) {
            const float f = __builtin_amdgcn_exp2f(acc[v]);
            fs[t][v] = f;
            tr[(laneHi * 8 + v) * 17 + laneLo] = f;   // [m][r_in_tile]
        }
        asm volatile("s_wait_dscnt 0x0" ::: "memory");  // LDS in-order fence

        // ---- GEMM2: oacc += fs_tile @ scons_ext_tile (4 K-slices of 4) ----
#pragma unroll
        for (int ks = 0; ks < 4; ++ks) {
            const int k0 = ks * 4 + laneHi * 2;       // r within tile
            v2f af, bf;
            af.x = tr[laneLo * 17 + k0 + 0];
            af.y = tr[laneLo * 17 + k0 + 1];
            bf.x = scons_ext[(rBase + k0 + 0) * CE + laneLo];
            bf.y = scons_ext[(rBase + k0 + 1) * CE + laneLo];
            oacc = __builtin_amdgcn_wmma_f32_16x16x4_f32(
                false, af, false, bf, (short)0, oacc, false, false);
        }
    }

    // ---- Reduce GEMM2 partials across the 16 waves (D-layout: col=laneLo) -
#pragma unroll
    for (int v = 0; v < 8; ++v)
        atomicAdd(&out_un[laneHi * 8 + v][laneLo], oacc[v]);   // ds_add_f32
    __syncthreads();

    // ---- Normalize in registers; single store of norm_fs ------------------
    // Row sums live in the ones-column (col C) of out_un.
    float scale[8];
#pragma unroll
    for (int v = 0; v < 8; ++v)
        scale[v] = 1.0f / (out_un[laneHi * 8 + v][C] + EPS);

#pragma unroll
    for (int t = 0; t < 8; ++t) {
        const int r = (wave * 8 + t) * 16 + laneLo;
#pragma unroll
        for (int v = 0; v < 8; ++v) {
            const int row = laneHi * 8 + v;
            norm_fs[(size_t)(bBase + row) * R + r] = fs[t][v] * scale[v];
        }
    }

    // ---- Final output: out[b,c] = s_x[b] * out_un[m,c] / (rowsum + eps) ---
    if (tid < 16 * C) {
        const int m2 = tid / C, c2 = tid % C;
        out[(bBase + m2) * C + c2] =
            sxs[m2] * out_un[m2][c2] / (out_un[m2][C] + EPS);
    }
}

// ---------------------------------------------------------------------------
extern "C" void kernel_launch(void* const* d_in, const int* in_sizes, int n_in,
                              void* d_out, int out_size, void* d_ws, size_t ws_size,
                              hipStream_t stream) {
    const float* x           = (const float*)d_in[0];   // (B, D)
    const float* centers     = (const float*)d_in[1];   // (D, M)
    const float* widths      = (const float*)d_in[2];   // (D, M)
    const float* consequents = (const float*)d_in[3];   // (R, D+1, C)
    const int*   rules       = (const int*)  d_in[4];   // (R, D)

    float* out     = (float*)d_out;                     // (B, C)
    float* norm_fs = out + (size_t)B * C;               // (B, R)
    float* x_ext   = norm_fs + (size_t)B * R;           // (B, D+1)

    // Workspace: 512 KB selector pack + 128 KB extended consequent sums.
    float* sel_pack  = (float*)d_ws;
    float* scons_ext = sel_pack + SELPACK_FLOATS;

    const int prep_threads = SELPACK_FLOATS;            // largest section
    anfis_prep_kernel<<<(prep_threads + 255) / 256, 256, 0, stream>>>(
        x, consequents, rules, sel_pack, scons_ext, x_ext);

    anfis_main_kernel<<<B / 16, 512, 0, stream>>>(
        x, centers, widths, sel_pack, scons_ext, out, norm_fs);
}